// RoIAlign_74148315398787
// MI455X (gfx1250) — compile-verified
//
#include <hip/hip_runtime.h>
#include <stdint.h>

// Problem constants (fixed by the reference setup)
#define FM_N   4
#define FM_C   256
#define FM_H   200
#define FM_W   200
#define NBOX   512
#define CROP   14
#define NPTS   (CROP * CROP)        // 196 sample points per box
#define TBL_DW (8 * NPTS)           // per box: 196 float4 weights + 196 int4 offsets
#define HWSZ   (FM_H * FM_W)        // 40000
#define CHW    ((size_t)FM_C * HWSZ)
#define CH_PER_BLK 64
#define CBLKS  (FM_C / CH_PER_BLK)  // 4 channel blocks per box
#define BLK    256                  // 8 wave32s per workgroup

typedef int v4i __attribute__((ext_vector_type(4)));
typedef __attribute__((address_space(1))) v4i gv4i;  // global (prints as __device__)
typedef __attribute__((address_space(3))) v4i lv4i;  // LDS

// ---- CDNA5 async global->LDS copy (ASYNCcnt path) -------------------------
__device__ __forceinline__ void async_copy_b128(const uint32_t* g, uint32_t* l) {
#if __has_builtin(__builtin_amdgcn_global_load_async_to_lds_b128)
  __builtin_amdgcn_global_load_async_to_lds_b128(
      (gv4i*)(void*)g, (lv4i*)(void*)l, 0, 0);
#else
  asm volatile("global_load_async_to_lds_b128 %0, %1, off"
               :: "v"((uint32_t)(uintptr_t)l), "v"((uint64_t)(uintptr_t)g)
               : "memory");
#endif
}

__device__ __forceinline__ void async_wait0() {
#if __has_builtin(__builtin_amdgcn_s_wait_asynccnt)
  __builtin_amdgcn_s_wait_asynccnt(0);
#else
  asm volatile("s_wait_asynccnt 0x0" ::: "memory");
#endif
}

// ---- Per-sample-point weights/offsets (mirrors the reference math) --------
__device__ __forceinline__ void roi_entry(const float4 box, int p,
    float4& w, int4& o)
{
  const int i = p / CROP;
  const int j = p - i * CROP;
  const float x1 = box.x, y1v = box.y, x2 = box.z, y2v = box.w;
  const float spw = (x2 - x1) / (float)CROP;
  const float sph = (y2v - y1v) / (float)CROP;
  const float nx0 = (x1 + spw * 0.5f - 0.5f) / (float)(FM_W - 1);
  const float ny0 = (y1v + sph * 0.5f - 0.5f) / (float)(FM_H - 1);
  const float nw  = spw * (float)(CROP - 1) / (float)(FM_W - 1);
  const float nh  = sph * (float)(CROP - 1) / (float)(FM_H - 1);
  const float gy  = (float)i / (float)(CROP - 1);
  const float gx  = (float)j / (float)(CROP - 1);
  const float py  = (ny0 + nh * gy) * (float)(FM_H - 1);
  const float px  = (nx0 + nw * gx) * (float)(FM_W - 1);
  const float y0f = floorf(py), x0f = floorf(px);
  const float wy = py - y0f, wx = px - x0f;
  const int y0 = (int)y0f, x0 = (int)x0f;
  const int y1i = y0 + 1, x1i = x0 + 1;

  const float vy0 = (y0  >= 0 && y0  < FM_H) ? 1.0f : 0.0f;
  const float vy1 = (y1i >= 0 && y1i < FM_H) ? 1.0f : 0.0f;
  const float vx0 = (x0  >= 0 && x0  < FM_W) ? 1.0f : 0.0f;
  const float vx1 = (x1i >= 0 && x1i < FM_W) ? 1.0f : 0.0f;

  w.x = (1.0f - wy) * (1.0f - wx) * (vy0 * vx0);
  w.y = (1.0f - wy) * wx          * (vy0 * vx1);
  w.z = wy          * (1.0f - wx) * (vy1 * vx0);
  w.w = wy          * wx          * (vy1 * vx1);

  const int yc0 = y0  < 0 ? 0 : (y0  > FM_H - 1 ? FM_H - 1 : y0);
  const int yc1 = y1i < 0 ? 0 : (y1i > FM_H - 1 ? FM_H - 1 : y1i);
  const int xc0 = x0  < 0 ? 0 : (x0  > FM_W - 1 ? FM_W - 1 : x0);
  const int xc1 = x1i < 0 ? 0 : (x1i > FM_W - 1 ? FM_W - 1 : x1i);

  o.x = yc0 * FM_W + xc0;
  o.y = yc0 * FM_W + xc1;
  o.z = yc1 * FM_W + xc0;
  o.w = yc1 * FM_W + xc1;
}

// ---- Kernel 1: build the sample table once (avoids 256x VALU recompute) ---
// Per-box chunk layout (contiguous 6272B): [196 x float4 weights][196 x int4 offs]
__global__ void __launch_bounds__(BLK)
roi_build_table(const float* __restrict__ boxes, uint32_t* __restrict__ tbl)
{
  const int tid = blockIdx.x * blockDim.x + threadIdx.x;
  if (tid >= NBOX * NPTS) return;
  const int m = tid / NPTS;
  const int p = tid - m * NPTS;
  const float4 box = reinterpret_cast<const float4*>(boxes)[m];
  float4 w; int4 o;
  roi_entry(box, p, w, o);
  uint32_t* chunk = tbl + (size_t)m * TBL_DW;
  reinterpret_cast<float4*>(chunk)[p]            = w;   // global_store_b128
  reinterpret_cast<int4*>(chunk + 4 * NPTS)[p]   = o;   // global_store_b128
}

// ---- Shared gather body ---------------------------------------------------
__device__ __forceinline__ void gather_body(
    const uint32_t* s_tbl, const float* __restrict__ fm,
    const int* __restrict__ box_ind, float* __restrict__ out,
    int m, int c0, int tid)
{
  const float4* s_w = reinterpret_cast<const float4*>(s_tbl);
  const int4*   s_o = reinterpret_cast<const int4*>(s_tbl + 4 * NPTS);

  const int n = box_ind[m];
  const float* base = fm + (size_t)n * CHW + (size_t)c0 * HWSZ;
  float* ob = out + ((size_t)m * FM_C + c0) * NPTS;

  for (int t = tid; t < CH_PER_BLK * NPTS; t += BLK) {
    const int c = t / NPTS;
    const int p = t - c * NPTS;
    const float* plane = base + (size_t)c * HWSZ;
    const float4 w = s_w[p];   // ds_load_b128, conflict-free (16B lane stride)
    const int4   o = s_o[p];   // ds_load_b128
    const float v = w.x * plane[o.x] + w.y * plane[o.y]
                  + w.z * plane[o.z] + w.w * plane[o.w];
    // NT store: don't let the 103MB output stream evict the L2-resident featuremap
    __builtin_nontemporal_store(v, &ob[t]);
  }
}

// ---- Kernel 2: async-stage table to LDS, gather 64 channels x 196 pts -----
__global__ void __launch_bounds__(BLK)
roi_gather(const float* __restrict__ fm, const int* __restrict__ box_ind,
           const uint32_t* __restrict__ tbl, float* __restrict__ out)
{
  __shared__ uint32_t s_tbl[TBL_DW];
  const int bx  = blockIdx.x;
  const int m   = bx / CBLKS;
  const int c0  = (bx - m * CBLKS) * CH_PER_BLK;
  const int tid = threadIdx.x;

  // Stage this box's 6272B table: 392 async b128 transfers (ASYNCcnt path)
  const uint32_t* gsrc = tbl + (size_t)m * TBL_DW;
  for (int k = tid; k < TBL_DW / 4; k += BLK)
    async_copy_b128(gsrc + 4 * k, &s_tbl[4 * k]);
  async_wait0();
  __syncthreads();

  gather_body(s_tbl, fm, box_ind, out, m, c0, tid);
}

// ---- Fallback: fused single kernel (no workspace needed) ------------------
__global__ void __launch_bounds__(BLK)
roi_fused(const float* __restrict__ fm, const float* __restrict__ boxes,
          const int* __restrict__ box_ind, float* __restrict__ out)
{
  __shared__ uint32_t s_tbl[TBL_DW];
  const int bx  = blockIdx.x;
  const int m   = bx / CBLKS;
  const int c0  = (bx - m * CBLKS) * CH_PER_BLK;
  const int tid = threadIdx.x;

  if (tid < NPTS) {
    const float4 box = reinterpret_cast<const float4*>(boxes)[m];
    float4 w; int4 o;
    roi_entry(box, tid, w, o);
    reinterpret_cast<float4*>(s_tbl)[tid]          = w;  // ds_store_b128
    reinterpret_cast<int4*>(s_tbl + 4 * NPTS)[tid] = o;  // ds_store_b128
  }
  __syncthreads();

  gather_body(s_tbl, fm, box_ind, out, m, c0, tid);
}

extern "C" void kernel_launch(void* const* d_in, const int* in_sizes, int n_in,
                              void* d_out, int out_size, void* d_ws, size_t ws_size,
                              hipStream_t stream) {
  const float* fm      = (const float*)d_in[0];
  const float* boxes   = (const float*)d_in[1];
  const int*   box_ind = (const int*)d_in[2];
  float*       out     = (float*)d_out;

  const size_t tbl_bytes = (size_t)NBOX * TBL_DW * sizeof(uint32_t); // ~3.2 MB
  if (ws_size >= tbl_bytes) {
    uint32_t* tbl = (uint32_t*)d_ws;
    roi_build_table<<<(NBOX * NPTS + BLK - 1) / BLK, BLK, 0, stream>>>(boxes, tbl);
    roi_gather<<<NBOX * CBLKS, BLK, 0, stream>>>(fm, box_ind, tbl, out);
  } else {
    roi_fused<<<NBOX * CBLKS, BLK, 0, stream>>>(fm, boxes, box_ind, out);
  }
}